// PrototypeContrastLoss_22539988369558
// MI455X (gfx1250) — compile-verified
//
#include <hip/hip_runtime.h>

// PrototypeContrastLoss on gfx1250 (MI455X).
// Bandwidth-bound: 288 MB streamed once -> ~12.4 us floor at 23.3 TB/s.
// b128 coalesced loads, fp32 accumulation, one block per row.
// Cross-wave block reduction done with V_WMMA_F32_16X16X4_F32 (fp32-exact).

typedef float v8f __attribute__((ext_vector_type(8)));
typedef float v2f __attribute__((ext_vector_type(2)));

constexpr int BS = 1024;
constexpr int D  = 4096;
constexpr int K  = 8;
constexpr int THREADS = 256;                       // 8 waves (wave32)
constexpr int F4_PER_ROW = D / 4;                  // 1024 float4 per row
constexpr int F4_PER_THREAD = F4_PER_ROW / THREADS; // 4

__device__ __forceinline__ float wave_bfly_sum(float v) {
  #pragma unroll
  for (int off = 16; off > 0; off >>= 1)
    v += __shfl_xor(v, off, 32);
  return v;   // all 32 lanes hold the wave total
}

__global__ __launch_bounds__(THREADS)
void row_loss_kernel(const float* __restrict__ base,
                     const float* __restrict__ pos,
                     const float* __restrict__ cross,
                     const float* __restrict__ neg,
                     float* __restrict__ row_loss) {
  const int i = blockIdx.x;     // row
  const int t = threadIdx.x;
  const int lane = t & 31;
  const int wave = t >> 5;

  // 8 waves x 16 slots (7 used, rest must be zero for the WMMA ones-trick)
  __shared__ float red[8 * 16];
  if (t < 128) red[t] = 0.0f;
  __syncthreads();

  const float4* b4 = reinterpret_cast<const float4*>(base + (size_t)i * D);
  const float4* g4 = reinterpret_cast<const float4*>(neg  + (size_t)i * D);

  float s1 = 0.f, s2 = 0.f, sn = 0.f, nb = 0.f, n1 = 0.f, n2 = 0.f, nn = 0.f;

  #pragma unroll 1   // keep ~18 b128 loads in flight per iter; preserve occupancy
  for (int j = 0; j < F4_PER_THREAD; ++j) {
    const int idx = j * THREADS + t;               // coalesced: wave reads 512B
    const float4 bb = b4[idx];
    const float4 gg = g4[idx];

    float4 p1 = make_float4(0.f, 0.f, 0.f, 0.f);
    float4 p2 = make_float4(0.f, 0.f, 0.f, 0.f);
    #pragma unroll
    for (int k = 0; k < K; ++k) {
      const float4* pk = reinterpret_cast<const float4*>(pos + ((size_t)k * BS + i) * D);
      const float4 v = pk[idx];
      p1.x += v.x; p1.y += v.y; p1.z += v.z; p1.w += v.w;
    }
    #pragma unroll
    for (int k = 0; k < K; ++k) {
      const float4* ck = reinterpret_cast<const float4*>(cross + ((size_t)k * BS + i) * D);
      const float4 v = ck[idx];
      p2.x += v.x; p2.y += v.y; p2.z += v.z; p2.w += v.w;
    }

    s1 += bb.x*p1.x + bb.y*p1.y + bb.z*p1.z + bb.w*p1.w;
    s2 += bb.x*p2.x + bb.y*p2.y + bb.z*p2.z + bb.w*p2.w;
    sn += bb.x*gg.x + bb.y*gg.y + bb.z*gg.z + bb.w*gg.w;
    nb += bb.x*bb.x + bb.y*bb.y + bb.z*bb.z + bb.w*bb.w;
    n1 += p1.x*p1.x + p1.y*p1.y + p1.z*p1.z + p1.w*p1.w;
    n2 += p2.x*p2.x + p2.y*p2.y + p2.z*p2.z + p2.w*p2.w;
    nn += gg.x*gg.x + gg.y*gg.y + gg.z*gg.z + gg.w*gg.w;
  }

  // 32-lane exact fp32 butterfly per quantity
  s1 = wave_bfly_sum(s1); s2 = wave_bfly_sum(s2); sn = wave_bfly_sum(sn);
  nb = wave_bfly_sum(nb); n1 = wave_bfly_sum(n1); n2 = wave_bfly_sum(n2);
  nn = wave_bfly_sum(nn);

  if (lane == 0) {
    float* r = &red[wave * 16];
    r[0] = s1; r[1] = s2; r[2] = sn; r[3] = nb; r[4] = n1; r[5] = n2; r[6] = nn;
  }
  __syncthreads();

  // Collapse the 8x16 wave-partial tile with two accumulating fp32 WMMAs.
  // A = ones(16x4) => C[m][n] = sum_k B[k][n].  B layout (4x16, 2 VGPRs):
  //   B reg0: lanes 0-15 = row0[n], lanes 16-31 = row1[n]
  //   B reg1: lanes 0-15 = row2[n], lanes 16-31 = row3[n]
  // Rows 0..3 <- waves 0..3 (first WMMA), rows <- waves 4..7 (second, C-acc).
  // All 8 waves execute identically (EXEC all-ones requirement for WMMA).
  const int col  = lane & 15;
  const int half = (lane >> 4) & 1;
  v2f a;  a.x = 1.0f;                      a.y = 1.0f;
  v2f b;  b.x = red[(0 + half) * 16 + col]; b.y = red[(2 + half) * 16 + col];
  v2f b2; b2.x = red[(4 + half) * 16 + col]; b2.y = red[(6 + half) * 16 + col];
  v8f c = {};
  c = __builtin_amdgcn_wmma_f32_16x16x4_f32(false, a, false, b,  (short)0, c, false, false);
  c = __builtin_amdgcn_wmma_f32_16x16x4_f32(false, a, false, b2, (short)0, c, false, false);

  // c[0], lane n (n<16) = block total for quantity n (broadcast across rows).
  const float cv = c[0];
  const float S0 = __shfl(cv, 0, 32);
  const float S1 = __shfl(cv, 1, 32);
  const float S2 = __shfl(cv, 2, 32);
  const float S3 = __shfl(cv, 3, 32);
  const float S4 = __shfl(cv, 4, 32);
  const float S5 = __shfl(cv, 5, 32);
  const float S6 = __shfl(cv, 6, 32);

  if (t == 0) {
    const float EPS = 1e-8f;
    // mean over K applied post-hoc: dot/8, norm = sqrt(sumsq)/8
    const float dot1 = S0 * 0.125f;
    const float dot2 = S1 * 0.125f;
    const float dotn = S2;
    const float na  = fmaxf(sqrtf(S3), EPS);
    const float nm1 = fmaxf(sqrtf(S4) * 0.125f, EPS);
    const float nm2 = fmaxf(sqrtf(S5) * 0.125f, EPS);
    const float nmn = fmaxf(sqrtf(S6), EPS);
    const float c1v = dot1 / (na * nm1);
    const float c2v = dot2 / (na * nm2);
    const float cnv = dotn / (na * nmn);
    const float pl = expf(c1v) + expf(c2v);
    const float nl = expf(cnv) + pl;
    row_loss[i] = -logf(pl / nl + 1e-8f);
  }
}

__global__ __launch_bounds__(THREADS)
void finalize_kernel(const float* __restrict__ row_loss, float* __restrict__ out) {
  const int t = threadIdx.x;
  float v = row_loss[t] + row_loss[t + 256] + row_loss[t + 512] + row_loss[t + 768];
  v = wave_bfly_sum(v);
  __shared__ float acc[8];
  if ((t & 31) == 0) acc[t >> 5] = v;
  __syncthreads();
  if (t == 0) {
    float s = 0.f;
    #pragma unroll
    for (int w = 0; w < 8; ++w) s += acc[w];
    out[0] = s * (1.0f / (float)BS);
  }
}

extern "C" void kernel_launch(void* const* d_in, const int* in_sizes, int n_in,
                              void* d_out, int out_size, void* d_ws, size_t ws_size,
                              hipStream_t stream) {
  (void)in_sizes; (void)n_in; (void)out_size; (void)ws_size;
  const float* base  = (const float*)d_in[0];
  const float* pos   = (const float*)d_in[1];
  const float* cross = (const float*)d_in[2];
  const float* neg   = (const float*)d_in[3];
  float* row_loss = (float*)d_ws;   // 1024 floats of scratch

  row_loss_kernel<<<BS, THREADS, 0, stream>>>(base, pos, cross, neg, row_loss);
  finalize_kernel<<<1, THREADS, 0, stream>>>(row_loss, (float*)d_out);
}